// Set_halo_84507776516392
// MI455X (gfx1250) — compile-verified
//
#include <hip/hip_runtime.h>
#include <hip/hip_bf16.h>

// Shape: [1, 8, 8, 8, 258, 258, 1] -> 512 independent 258x258 fp32 images.
// Pure streaming problem: 409 MB moved, ~34 MFLOP -> bandwidth-bound on the
// 23.3 TB/s HBM path (~17.6 us ideal). Working set > 192 MB L2, so use
// non-temporal 128-bit global accesses (CDNA5 TH_NT cache policy) and wave32
// blocks. No matrix math -> no WMMA by design.

typedef float v4f __attribute__((ext_vector_type(4)));

#define HW   258
#define IMG  (258 * 258)        // 66564 elements per image
#define NBRD 1028               // exact border element count per image

// ---------------------------------------------------------------------------
// Bulk: out = x * halo over the whole tensor, 1 float4 per lane, NT streaming.
// ---------------------------------------------------------------------------
__global__ void __launch_bounds__(256)
halo_bulk_mul(const v4f* __restrict__ x,
              const v4f* __restrict__ hm,
              v4f* __restrict__ out,
              int n4) {
    int i = blockIdx.x * blockDim.x + threadIdx.x;
    if (i >= n4) return;
    v4f a = __builtin_nontemporal_load(&x[i]);    // global_load_b128 th:NT
    v4f b = __builtin_nontemporal_load(&hm[i]);
    __builtin_nontemporal_store(a * b, &out[i]);  // global_store_b128 th:NT
}

// ---------------------------------------------------------------------------
// Border fixup: each of the 1028 border elements per image written exactly
// once, with the reference's precedence (W-axis assigns win at corners).
//   t in [0,256)    : h=0,    w=t+1      src = x[1][w]
//   t in [256,512)  : h=257,  w=t-255    src = x[256][w]
//   t in [512,770)  : h=t-512, w=0       src = x[h][1]     (covers corners)
//   t in [770,1028) : h=t-770, w=257     src = x[h][256]   (covers corners)
// ---------------------------------------------------------------------------
__global__ void __launch_bounds__(256)
halo_border_fix(const float* __restrict__ x,
                const float* __restrict__ hm,
                float* __restrict__ out) {
    int t = blockIdx.x * blockDim.x + threadIdx.x;
    if (t >= NBRD) return;
    size_t base = (size_t)blockIdx.y * IMG;
    const float* xi = x   + base;
    const float* hi = hm  + base;
    float*       oi = out + base;

    int h, w, sh, sw;
    if (t < 256)      { h = 0;       w = t + 1;   sh = 1;   sw = w;   }
    else if (t < 512) { h = HW - 1;  w = t - 255; sh = 256; sw = w;   }
    else if (t < 770) { h = t - 512; w = 0;       sh = h;   sw = 1;   }
    else              { h = t - 770; w = HW - 1;  sh = h;   sw = 256; }

    int pos = h * HW + w;
    oi[pos] = xi[sh * HW + sw] * hi[pos];
}

extern "C" void kernel_launch(void* const* d_in, const int* in_sizes, int n_in,
                              void* d_out, int out_size, void* d_ws, size_t ws_size,
                              hipStream_t stream) {
    const float* x  = (const float*)d_in[0];
    const float* hm = (const float*)d_in[1];
    float* out      = (float*)d_out;

    const int total = in_sizes[0];          // 34,080,768 (divisible by 4)
    const int n4    = total / 4;
    const int nimg  = total / IMG;          // 512

    // Bulk streaming multiply: 256 threads = 8 wave32 per block.
    int blocks = (n4 + 255) / 256;
    halo_bulk_mul<<<blocks, 256, 0, stream>>>(
        (const v4f*)x, (const v4f*)hm, (v4f*)out, n4);

    // Border fixup (ordered after bulk on the same stream).
    dim3 fgrid((NBRD + 255) / 256, nimg);
    halo_border_fix<<<fgrid, 256, 0, stream>>>(x, hm, out);
}